// GCN_8297876816011
// MI455X (gfx1250) — compile-verified
//
#include <hip/hip_runtime.h>

// ---------------------------------------------------------------------------
// GCN forward for MI455X (gfx1250, wave32).
//   out = D^-1/2 A D^-1/2 relu(D^-1/2 A D^-1/2 (X@W1)) @ W2
// Decomposition used:  D^-1/2 A D^-1/2 Y  =  D^-1/2 * (A * (D^-1/2 * Y))
//   - GEMM epilogue scales C rows by dinv  (the "D^-1/2 * Y" part)
//   - plain gather/atomic-scatter SpMM     (the "A *" part)
//   - next GEMM's A-load applies dinv (+ReLU) (the leading "D^-1/2 *")
// GEMMs: V_WMMA_F32_16X16X4_F32 (exact f32), B staged in LDS (K-paired layout
// so B fragments are single ds_load_b64), A fragment reused across all N tiles.
// ---------------------------------------------------------------------------

typedef float v2f __attribute__((ext_vector_type(2)));
typedef float v8f __attribute__((ext_vector_type(8)));

#define GCN_N_NODES 100000

// ---------------- utility kernels ----------------

__global__ void gcn_zero_kernel(float* __restrict__ p, long long n) {
  long long i = (long long)blockIdx.x * blockDim.x + threadIdx.x;
  long long stride = (long long)gridDim.x * blockDim.x;
  for (; i < n; i += stride) p[i] = 0.0f;
}

__global__ void gcn_degree_kernel(const int* __restrict__ rows,
                                  float* __restrict__ deg, int n_edges) {
  int e = blockIdx.x * blockDim.x + threadIdx.x;
  if (e < n_edges) atomicAdd(&deg[rows[e]], 1.0f);
}

__global__ void gcn_dinv_kernel(const float* __restrict__ deg,
                                float* __restrict__ dinv, int n) {
  int i = blockIdx.x * blockDim.x + threadIdx.x;
  if (i < n) {
    float d = sqrtf(deg[i]);
    dinv[i] = 1.0f / fmaxf(d, 1e-12f);
  }
}

// out[i] *= dinv[i >> colshift]   (final row scaling of the output)
__global__ void gcn_rowscale_kernel(float* __restrict__ p,
                                    const float* __restrict__ dinv,
                                    long long n, int colshift) {
  long long i = (long long)blockIdx.x * blockDim.x + threadIdx.x;
  long long stride = (long long)gridDim.x * blockDim.x;
  for (; i < n; i += stride) p[i] *= dinv[i >> colshift];
}

// ---------------- WMMA f32 GEMM: C[MxN] = op(A)[MxK] @ B[KxN] -------------
// One wave -> 16 rows x full N (NT = N/16 accumulator tiles; A reused NT x).
// Block = 8 waves = 128 rows. B staged per 64-K chunk into LDS, layout
// Bs[kpair][col][2] with element (kp,col,j) = B[2*kp+j][col] so each WMMA B
// fragment (lane: {B[k][col], B[k+1][col]}) is one contiguous ds_load_b64.
// ISA VGPR layouts (CDNA5 ISA §7.12.2):
//   A 16x4:  lanes 0-15: M=lane, {K=0,1}; lanes 16-31: M=lane-16, {K=2,3}
//   B 4x16:  VGPR v, lanes 0-15: K=v, N=lane; lanes 16-31: K=v+2, N=lane-16
//   C 16x16: VGPR v, lanes 0-15: M=v, N=lane; lanes 16-31: M=v+8, N=lane-16
// RELU_SCALE_A: A element -> relu(dinv[row] * a).  SCALE_C: C row *= dinv[row].

template <int N, bool RELU_SCALE_A, bool SCALE_C>
__global__ __launch_bounds__(256) void gcn_gemm_wmma(
    const float* __restrict__ A, const float* __restrict__ B,
    float* __restrict__ C, const float* __restrict__ dinv, int M, int K) {
  constexpr int NT = N / 16;   // 8 (N=128) or 4 (N=64)
  constexpr int KC = 64;       // K-chunk staged in LDS
  constexpr int N2 = N / 2;
  __shared__ __align__(16) float Bs[KC * N];  // 32KB / 16KB

  const int wib  = threadIdx.x >> 5;          // wave in block: 0..7
  const int lane = threadIdx.x & 31;
  const int half = lane >> 4;                 // selects K pair {0,1} vs {2,3}
  const int l15  = lane & 15;

  const int row0   = blockIdx.x * 128 + wib * 16;  // wave's first row
  const bool active = (row0 < M);                  // wave-uniform guard

  const float* __restrict__ Arow =
      A + (size_t)(active ? row0 + l15 : 0) * K;
  const float dscale =
      (RELU_SCALE_A && active) ? dinv[row0 + l15] : 1.0f;

  v8f acc[NT];
#pragma unroll
  for (int t = 0; t < NT; ++t) acc[t] = (v8f){};

  for (int kc = 0; kc < K; kc += KC) {
    // ---- cooperative B staging: all 256 threads, K-paired layout ----
    const float* __restrict__ Bg = B + (size_t)kc * N;
#pragma unroll
    for (int pidx = threadIdx.x; pidx < (KC / 2) * N2; pidx += 256) {
      const int kp = pidx / N2;
      const int c2 = (pidx - kp * N2) * 2;
      const float2 g0 = *(const float2*)(Bg + (size_t)(2 * kp) * N + c2);
      const float2 g1 = *(const float2*)(Bg + (size_t)(2 * kp + 1) * N + c2);
      *(float4*)(&Bs[kp * (2 * N) + c2 * 2]) =
          make_float4(g0.x, g1.x, g0.y, g1.y);      // ds_store_b128
    }
    __syncthreads();

    if (active) {
      const float* __restrict__ Ak = Arow + kc;
#pragma unroll
      for (int k0 = 0; k0 < KC; k0 += 4) {
        const int koff = k0 + 2 * half;
        v2f a = *(const v2f*)(Ak + koff);           // global_load_b64
        if (RELU_SCALE_A) {
          a.x = fmaxf(a.x * dscale, 0.0f);
          a.y = fmaxf(a.y * dscale, 0.0f);
        }
        const int kp = koff >> 1;
#pragma unroll
        for (int t = 0; t < NT; ++t) {
          const v2f b =
              *(const v2f*)(&Bs[kp * (2 * N) + (t * 16 + l15) * 2]);  // ds_load_b64
          acc[t] = __builtin_amdgcn_wmma_f32_16x16x4_f32(
              false, a, false, b, (short)0, acc[t], false, false);
        }
      }
    }
    __syncthreads();
  }

  if (active) {
#pragma unroll
    for (int t = 0; t < NT; ++t) {
      const int col = t * 16 + l15;
#pragma unroll
      for (int v = 0; v < 8; ++v) {
        const int row = row0 + v + 8 * half;
        float val = acc[t][v];
        if (SCALE_C) val *= dinv[row];
        C[(size_t)row * N + col] = val;
      }
    }
  }
}

// ---------------- plain SpMM: dst[row] += src[col] (per edge) -------------
// F/4 float4 chunks per edge; fshift = log2(F/4) so decode is shift/mask.

__global__ void gcn_spmm_plain(const float* __restrict__ src,
                               const int* __restrict__ rows,
                               const int* __restrict__ cols,
                               float* __restrict__ dst,
                               int n_edges, int F, int fshift) {
  const long long tid = (long long)blockIdx.x * blockDim.x + threadIdx.x;
  const int fchunks = 1 << fshift;
  const long long total = (long long)n_edges * fchunks;
  if (tid >= total) return;

  const int e  = (int)(tid >> fshift);
  const int fc = (int)(tid & (fchunks - 1));

  const int r = rows[e];
  const int c = cols[e];

  const float4 v = ((const float4*)(src + (size_t)c * F))[fc];
  float* __restrict__ d = dst + (size_t)r * F + fc * 4;
  atomicAdd(d + 0, v.x);
  atomicAdd(d + 1, v.y);
  atomicAdd(d + 2, v.z);
  atomicAdd(d + 3, v.w);
}

// ---------------- driver ----------------

extern "C" void kernel_launch(void* const* d_in, const int* in_sizes, int n_in,
                              void* d_out, int out_size, void* d_ws, size_t ws_size,
                              hipStream_t stream) {
  const float* X  = (const float*)d_in[0];   // [100000, 256]
  const float* W1 = (const float*)d_in[1];   // [256, 128]
  const float* W2 = (const float*)d_in[2];   // [128, 64]
  const int* erow = (const int*)d_in[3];     // [n_edges]
  const int* ecol = (const int*)d_in[4];     // [n_edges]
  const int n_edges = in_sizes[3];
  float* out = (float*)d_out;                // [100000, 64]

  // workspace layout (floats)
  float* deg = (float*)d_ws;                         // 100000
  float* dinv = deg + GCN_N_NODES;                   // 100000
  float* XWs = dinv + GCN_N_NODES;                   // dinv-prescaled X@W1
  float* S   = XWs + (size_t)GCN_N_NODES * 128;      // SpMM1 accumulator
  float* HWs = S + (size_t)GCN_N_NODES * 128;        // dinv-prescaled H@W2

  const long long nS   = (long long)GCN_N_NODES * 128;
  const long long nOut = (long long)out_size;        // 100000*64

  // zero atomic destinations (ws/out are poisoned by harness)
  gcn_zero_kernel<<<512, 256, 0, stream>>>(deg, GCN_N_NODES);
  gcn_zero_kernel<<<4096, 256, 0, stream>>>(S, nS);
  gcn_zero_kernel<<<4096, 256, 0, stream>>>(out, nOut);

  // degrees -> d^{-1/2}
  gcn_degree_kernel<<<(n_edges + 255) / 256, 256, 0, stream>>>(erow, deg, n_edges);
  gcn_dinv_kernel<<<(GCN_N_NODES + 255) / 256, 256, 0, stream>>>(deg, dinv, GCN_N_NODES);

  const int row_tiles = GCN_N_NODES / 16;            // 6250
  const int gemm_blocks = (row_tiles + 7) / 8;       // 8 waves/block

  // GEMM1: XWs = (X @ W1) * dinv[row]   (M=100000, N=128, K=256)
  gcn_gemm_wmma<128, false, true>
      <<<gemm_blocks, 256, 0, stream>>>(X, W1, XWs, dinv, GCN_N_NODES, 256);

  // SpMM1: S[r] += XWs[c]  over edges  (F=128 -> 32 chunks/edge)
  {
    const long long total = (long long)n_edges * 32;
    gcn_spmm_plain<<<(int)((total + 255) / 256), 256, 0, stream>>>(
        XWs, erow, ecol, S, n_edges, 128, 5);
  }

  // GEMM2: HWs = (relu(dinv[row]*S) @ W2) * dinv[row]  (M=100000, N=64, K=128)
  gcn_gemm_wmma<64, true, true>
      <<<gemm_blocks, 256, 0, stream>>>(S, W2, HWs, dinv, GCN_N_NODES, 128);

  // SpMM2: out[r] += HWs[c]  (F=64 -> 16 chunks/edge)
  {
    const long long total = (long long)n_edges * 16;
    gcn_spmm_plain<<<(int)((total + 255) / 256), 256, 0, stream>>>(
        HWs, erow, ecol, out, n_edges, 64, 4);
  }

  // final row scaling: out[r][:] *= dinv[r]   (64 cols -> shift 6)
  gcn_rowscale_kernel<<<4096, 256, 0, stream>>>(out, dinv, nOut, 6);
}